// Dilation1D_51625506898142
// MI455X (gfx1250) — compile-verified
//
#include <hip/hip_runtime.h>
#include <math.h>

// ---------------------------------------------------------------------------
// 1D morphological dilation (max-plus), 11-tap parabolic kernel, N = 2^24 f32.
// Memory-bound: 128 MB traffic -> ~5.7 us floor at 23.3 TB/s. Strategy:
// stage each input tile once via CDNA5 async global->LDS 128-bit loads
// (ASYNCcnt), compute the 11-tap max window out of LDS, nontemporal float4
// stores for the write-once output stream.
// ---------------------------------------------------------------------------

#define K_SIZE      11
#define HALF        5
#define BLOCK       256
#define PER_THREAD  4
#define TILE        (BLOCK * PER_THREAD)          // 1024 outputs per block
#define HALO_PAD    8                             // halo padded 5->8 for 16B alignment
#define LDS_FLOATS  (TILE + 2 * HALO_PAD)         // 1040 floats = 4160 B
#define CHUNKS      (LDS_FLOATS / 4)              // 260 x b128 async loads

#if defined(__has_builtin)
#  if __has_builtin(__builtin_amdgcn_global_load_async_to_lds_b128)
#    define HAS_ASYNC_LDS_BUILTIN 1
#  endif
#  if __has_builtin(__builtin_amdgcn_s_wait_asynccnt)
#    define HAS_WAIT_ASYNC_BUILTIN 1
#  endif
#endif

typedef int   v4i __attribute__((ext_vector_type(4)));
typedef float v4f __attribute__((ext_vector_type(4)));
typedef __attribute__((address_space(1))) v4i* gptr_v4i;   // global 128-bit chunk
typedef __attribute__((address_space(3))) v4i* lptr_v4i;   // LDS 128-bit chunk

__device__ __forceinline__ void async_copy_b128(const float* __restrict__ gsrc,
                                                float* lds_dst) {
#if defined(HAS_ASYNC_LDS_BUILTIN)
    __builtin_amdgcn_global_load_async_to_lds_b128(
        (gptr_v4i)(const void*)gsrc,
        (lptr_v4i)(void*)lds_dst,
        /*imm offset*/ 0, /*cpol*/ 0);
#else
    // vdst = 32-bit LDS byte address, vaddr = 64-bit global address
    unsigned lds_off = (unsigned)(unsigned long long)(void*)lds_dst;
    asm volatile("global_load_async_to_lds_b128 %0, %1, off"
                 :: "v"(lds_off), "v"(gsrc)
                 : "memory");
#endif
}

__device__ __forceinline__ void wait_async0() {
#if defined(HAS_WAIT_ASYNC_BUILTIN)
    __builtin_amdgcn_s_wait_asynccnt(0);
#else
    asm volatile("s_wait_asynccnt 0" ::: "memory");
#endif
}

__global__ __launch_bounds__(BLOCK)
void Dilation1D_51625506898142_kernel(const float* __restrict__ in,
                                      const float* __restrict__ scale,
                                      float* __restrict__ out,
                                      int n) {
    __shared__ __align__(16) float tile[LDS_FLOATS];

    const int tid    = threadIdx.x;
    const int start  = (int)blockIdx.x * TILE;    // first output of block (fits int: N=2^24)
    const int istart = start - HALO_PAD;          // first staged input (may be -8 for block 0)

    // Uniform-per-block branch: interior blocks take the async 128-bit path.
    const bool interior = (istart >= 0) && (start + TILE + HALO_PAD <= n);

    if (interior) {
        // 260 aligned b128 chunks: one per lane + 4 extra on lanes 0..3.
        async_copy_b128(in + istart + 4 * tid, &tile[4 * tid]);
        if (tid < (CHUNKS - BLOCK)) {
            const int c = tid + BLOCK;
            async_copy_b128(in + istart + 4 * c, &tile[4 * c]);
        }
        wait_async0();            // this wave's async LDS writes complete
    } else {
        // Boundary blocks: guarded loads, -inf pad (matches jnp.pad -inf).
        for (int idx = tid; idx < LDS_FLOATS; idx += BLOCK) {
            const int g = istart + idx;
            tile[idx] = (g >= 0 && g < n) ? in[g] : -INFINITY;
        }
    }
    __syncthreads();              // make all waves' LDS writes visible

    // h[k] = -(k-5)^2 / (4*scale): tiny uniform setup, exact division
    // (symmetric taps CSE to 6 unique divisions).
    const float s4 = 4.0f * scale[0];
    float h[K_SIZE];
#pragma unroll
    for (int k = 0; k < K_SIZE; ++k) {
        const float z = (float)(k - HALF);
        h[k] = -(z * z) / s4;
    }

    // Window for output t' = 4*tid + j lives at tile[t'+3 .. t'+13].
    const int base = PER_THREAD * tid + (HALO_PAD - HALF);   // 4*tid + 3
    float w[PER_THREAD + K_SIZE - 1];                        // 14 floats
#pragma unroll
    for (int j = 0; j < PER_THREAD + K_SIZE - 1; ++j)
        w[j] = tile[base + j];

    float r[PER_THREAD];
#pragma unroll
    for (int j = 0; j < PER_THREAD; ++j) {
        float acc = w[j] + h[0];
#pragma unroll
        for (int k = 1; k < K_SIZE; ++k)
            acc = fmaxf(acc, w[j + k] + h[k]);
        r[j] = acc;
    }

    const int o = start + PER_THREAD * tid;
    if (o + PER_THREAD <= n) {
        // Write-once output stream: nontemporal 128-bit store.
        v4f rv = { r[0], r[1], r[2], r[3] };
        __builtin_nontemporal_store(rv, (v4f*)(out + o));
    } else {
#pragma unroll
        for (int j = 0; j < PER_THREAD; ++j)
            if (o + j < n) out[o + j] = r[j];
    }
}

extern "C" void kernel_launch(void* const* d_in, const int* in_sizes, int n_in,
                              void* d_out, int out_size, void* d_ws, size_t ws_size,
                              hipStream_t stream) {
    const float* in    = (const float*)d_in[0];   // [N] f32 signal
    const float* scale = (const float*)d_in[1];   // [1] f32 scalar parameter
    float*       out   = (float*)d_out;           // [N] f32

    const int n      = in_sizes[0];
    const int blocks = (n + TILE - 1) / TILE;     // 16384 for N = 2^24

    Dilation1D_51625506898142_kernel<<<blocks, BLOCK, 0, stream>>>(in, scale, out, n);
}